// GeometricFlow_56160992362886
// MI455X (gfx1250) — compile-verified
//
#include <hip/hip_runtime.h>
#include <hip/hip_bf16.h>

// CDNA5 / gfx1250 fused GeometricFlow kernel.
// fp32 end-to-end, matrix work on V_WMMA_F32_16X16X4_F32 (wave32).
// Round 2: all WMMA operands staged via LDS, value-selects instead of
// predicated loads (no saveexec), rolled outer loops to keep VGPRs < 256.

typedef float v2f __attribute__((ext_vector_type(2)));
typedef float v8f __attribute__((ext_vector_type(8)));

#define M_DIM 8
#define HID   128
#define BATCH 4096

__device__ __forceinline__ float fast_tanh(float x) {
#if __has_builtin(__builtin_amdgcn_tanhf)
  return __builtin_amdgcn_tanhf(x);     // v_tanh_f32 on gfx1250
#else
  return tanhf(x);
#endif
}

__global__ __launch_bounds__(256)
void geoflow_kernel(const float* __restrict__ points,
                    const float* __restrict__ Wm1, const float* __restrict__ bm1,
                    const float* __restrict__ Wm2, const float* __restrict__ bm2,
                    const float* __restrict__ Wc1, const float* __restrict__ bc1,
                    const float* __restrict__ Wc2, const float* __restrict__ bc2,
                    const float* __restrict__ Wr1, const float* __restrict__ br1,
                    const float* __restrict__ Wr2, const float* __restrict__ br2,
                    const float* __restrict__ Wf1, const float* __restrict__ bf1,
                    const float* __restrict__ Wf2, const float* __restrict__ bf2,
                    float* __restrict__ out)
{
  __shared__ float s_p[8];           // points[b]
  __shared__ float s_mh[128];        // metric hidden (post-relu)
  __shared__ float s_g[64];          // metric g[i][j]
  __shared__ float s_base[128];      // bc1 + points @ Wc1[0:8]
  __shared__ float s_base2[256];     // br1 + points @ Wr1[0:8]
  __shared__ float s_chris[512];     // chris[i*64 + j*8 + k]
  __shared__ float s_rpart[8 * 64];  // per-wave Ricci partials
  __shared__ float s_ric[64];        // ric[i*8 + j]
  __shared__ float s_ricp[8];        // ric @ p
  __shared__ float s_fh[128];        // flow hidden
  // staged weights (keeps every WMMA-operand fetch in LDS -> no saveexec)
  __shared__ float s_wc1r[3 * 128];  // Wc1 rows 8..10
  __shared__ float s_wc2[128];       // Wc2
  __shared__ float s_wr1r[9 * 256];  // Wr1 rows 8..16
  __shared__ float s_wr2[256];       // Wr2

  const int tid  = threadIdx.x;
  const int lane = tid & 31;
  const int wave = tid >> 5;
  const int half = (lane < 16) ? 0 : 1;   // which 16-lane half of the wave
  const int l15  = lane & 15;
  const int b    = blockIdx.x;

  // ---- stage inputs / weights into LDS ----
  if (tid < 8) s_p[tid] = points[b * 8 + tid];
  #pragma unroll
  for (int r = 0; r < 2; ++r) {            // 384 floats
    int idx = tid + r * 256;
    if (idx < 3 * 128) s_wc1r[idx] = Wc1[8 * 128 + idx];
  }
  if (tid < 128) s_wc2[tid] = Wc2[tid];
  #pragma unroll
  for (int r = 0; r < 9; ++r) s_wr1r[r * 256 + tid] = Wr1[(8 + r) * 256 + tid];
  s_wr2[tid] = Wr2[tid];
  __syncthreads();

  // ---- per-batch bases (fold points + bias into one feature row) ----
  {
    float acc = br1[tid];                   // 256 threads, one each
    #pragma unroll
    for (int d = 0; d < 8; ++d) acc += s_p[d] * Wr1[d * 256 + tid];
    s_base2[tid] = acc;
  }
  if (tid < 128) {
    float mh = bm1[tid];
    float ba = bc1[tid];
    #pragma unroll
    for (int d = 0; d < 8; ++d) {
      float pv = s_p[d];
      mh += pv * Wm1[d * 128 + tid];
      ba += pv * Wc1[d * 128 + tid];
    }
    s_mh[tid]   = fmaxf(mh, 0.f);
    s_base[tid] = ba;
  }
  __syncthreads();

  // ---- metric output g = relu_hidden @ Wm2 + bm2 (tiny) ----
  if (tid < 64) {
    float acc = bm2[tid];
    for (int h = 0; h < 128; ++h) acc += s_mh[h] * Wm2[h * 64 + tid];
    s_g[tid] = acc;
  }
  __syncthreads();

  // =====================================================================
  // Christoffel: pre[h, t] = [Wc1[8,h], Wc1[9,h], Wc1[10,h], base[h]]
  //                          . [g_ij, g_jk, g_ki, 1]      (K = 4 exactly)
  // 8 hidden-tiles x 32 triple-tiles = 256 WMMAs / batch element.
  // Wave w owns triple-tiles w*4 .. w*4+3. B operands are ht-invariant.
  // =====================================================================
  {
    v2f Bv[4];                              // cached B per triple-tile
    #pragma unroll
    for (int tl = 0; tl < 4; ++tl) {
      const int t = (wave * 4 + tl) * 16 + l15;    // B: N = lane&15
      const int i = t >> 6, j = (t >> 3) & 7, k = t & 7;
      const float gij = s_g[i * 8 + j];
      const float gjk = s_g[j * 8 + k];
      const float gki = s_g[k * 8 + i];
      Bv[tl].x = half ? gki  : gij;         // K {2 | 0}
      Bv[tl].y = half ? 1.0f : gjk;         // K {3 | 1}
    }
    float acc[4] = {0.f, 0.f, 0.f, 0.f};
    const float bias2 = bc2[0];
    #pragma unroll 1
    for (int ht = 0; ht < 8; ++ht) {
      const int h = ht * 16 + l15;          // A: M = lane&15
      const float ax0 = s_wc1r[0 * 128 + h];
      const float ax1 = s_wc1r[2 * 128 + h];
      const float ay0 = s_wc1r[1 * 128 + h];
      const float ay1 = s_base[h];
      v2f A;
      A.x = half ? ax1 : ax0;
      A.y = half ? ay1 : ay0;
      float w[8];                           // Wc2 rows matching D layout M=r+8*half
      #pragma unroll
      for (int r = 0; r < 8; ++r) w[r] = s_wc2[ht * 16 + half * 8 + r];
      #pragma unroll
      for (int tl = 0; tl < 4; ++tl) {
        v8f D = {};
        D = __builtin_amdgcn_wmma_f32_16x16x4_f32(
                false, A, false, Bv[tl], (short)0, D, false, false);
        float part = 0.f;
        #pragma unroll
        for (int r = 0; r < 8; ++r) part += fast_tanh(D[r]) * w[r];
        acc[tl] += part;
      }
    }
    #pragma unroll
    for (int tl = 0; tl < 4; ++tl) {
      // lanes l and l+16 hold the two hidden halves of the same triple column
      float v = acc[tl] + __shfl_xor(acc[tl], 16, 32);
      if (lane < 16) s_chris[(wave * 4 + tl) * 16 + lane] = v + bias2;
    }
  }
  __syncthreads();

  // =====================================================================
  // Ricci: features per pair = [g_ij, chris[0..7], 1, 0, 0]  (K = 12)
  // 3 chained K=4 WMMAs per (16 hidden x 16 pairs) tile.
  // 16 hidden-tiles x 4 pair-tiles x 3 = 192 WMMAs / batch element.
  // Wave w owns hidden-tiles 2w, 2w+1; covers all 4 pair-tiles.
  // =====================================================================
  {
    float racc[4] = {0.f, 0.f, 0.f, 0.f};
    const int mt0 = wave * 2;
    v2f Ac[2][3];
    float w2[2][8];
    #pragma unroll
    for (int mi = 0; mi < 2; ++mi) {
      const int h2 = (mt0 + mi) * 16 + l15;
      // s_wr1r[f] = Wr1 row 8+f: f=0 -> g_ij row, f=1..8 -> chris rows
      Ac[mi][0].x = s_wr1r[(half ? 2 : 0) * 256 + h2];
      Ac[mi][0].y = s_wr1r[(half ? 3 : 1) * 256 + h2];
      Ac[mi][1].x = s_wr1r[(half ? 6 : 4) * 256 + h2];
      Ac[mi][1].y = s_wr1r[(half ? 7 : 5) * 256 + h2];
      const float a2x = s_wr1r[8 * 256 + h2];     // chris[7] weight row
      const float a2y = s_base2[h2];              // folded points+bias
      Ac[mi][2].x = half ? 0.f : a2x;             // K pad rows 10,11 are zero
      Ac[mi][2].y = half ? 0.f : a2y;
      #pragma unroll
      for (int r = 0; r < 8; ++r) w2[mi][r] = s_wr2[(mt0 + mi) * 16 + half * 8 + r];
    }
    #pragma unroll 1
    for (int nt = 0; nt < 4; ++nt) {
      const int pair = nt * 16 + l15;
      const int i = pair >> 3, j = pair & 7;
      const float gij = s_g[i * 8 + j];
      float c[8];
      #pragma unroll
      for (int k = 0; k < 8; ++k) c[k] = s_chris[pair * 8 + k];
      v2f Bc[3];
      Bc[0].x = half ? c[1] : gij;    // f 2 | 0
      Bc[0].y = half ? c[2] : c[0];   // f 3 | 1
      Bc[1].x = half ? c[5] : c[3];   // f 6 | 4
      Bc[1].y = half ? c[6] : c[4];   // f 7 | 5
      Bc[2].x = half ? 0.f  : c[7];   // f 10| 8
      Bc[2].y = half ? 0.f  : 1.0f;   // f 11| 9
      #pragma unroll
      for (int mi = 0; mi < 2; ++mi) {
        v8f D = {};
        #pragma unroll
        for (int cc = 0; cc < 3; ++cc)
          D = __builtin_amdgcn_wmma_f32_16x16x4_f32(
                  false, Ac[mi][cc], false, Bc[cc], (short)0, D, false, false);
        #pragma unroll
        for (int r = 0; r < 8; ++r) racc[nt] += fmaxf(D[r], 0.f) * w2[mi][r];
      }
    }
    #pragma unroll
    for (int nt = 0; nt < 4; ++nt) {
      float v = racc[nt] + __shfl_xor(racc[nt], 16, 32);
      if (lane < 16) s_rpart[wave * 64 + nt * 16 + lane] = v;
    }
  }
  __syncthreads();

  // reduce 8 wave-partials -> ric[i][j]
  if (tid < 64) {
    float acc = br2[0];
    #pragma unroll
    for (int w = 0; w < 8; ++w) acc += s_rpart[w * 64 + tid];
    s_ric[tid] = acc;
  }
  __syncthreads();

  // ---- flow network ----
  if (tid < 8) {
    float acc = 0.f;
    #pragma unroll
    for (int j = 0; j < 8; ++j) acc += s_ric[tid * 8 + j] * s_p[j];
    s_ricp[tid] = acc;
  }
  __syncthreads();
  if (tid < 128) {
    float acc = bf1[tid];
    #pragma unroll
    for (int d = 0; d < 8; ++d) acc += s_p[d]    * Wf1[d * 128 + tid];
    #pragma unroll
    for (int d = 0; d < 8; ++d) acc += s_ricp[d] * Wf1[(8 + d) * 128 + tid];
    s_fh[tid] = fmaxf(acc, 0.f);
  }
  __syncthreads();
  if (tid < 8) {
    float acc = bf2[tid];
    for (int h = 0; h < 128; ++h) acc += s_fh[h] * Wf2[h * 8 + tid];
    out[b * 8 + tid] = s_p[tid] + 0.01f * acc;
  }
}

extern "C" void kernel_launch(void* const* d_in, const int* in_sizes, int n_in,
                              void* d_out, int out_size, void* d_ws, size_t ws_size,
                              hipStream_t stream) {
  (void)in_sizes; (void)n_in; (void)out_size; (void)d_ws; (void)ws_size;
  const float* points = (const float*)d_in[0];
  const float* Wm1 = (const float*)d_in[1];
  const float* bm1 = (const float*)d_in[2];
  const float* Wm2 = (const float*)d_in[3];
  const float* bm2 = (const float*)d_in[4];
  const float* Wc1 = (const float*)d_in[5];
  const float* bc1 = (const float*)d_in[6];
  const float* Wc2 = (const float*)d_in[7];
  const float* bc2 = (const float*)d_in[8];
  const float* Wr1 = (const float*)d_in[9];
  const float* br1 = (const float*)d_in[10];
  const float* Wr2 = (const float*)d_in[11];
  const float* br2 = (const float*)d_in[12];
  const float* Wf1 = (const float*)d_in[13];
  const float* bf1 = (const float*)d_in[14];
  const float* Wf2 = (const float*)d_in[15];
  const float* bf2 = (const float*)d_in[16];
  float* outp = (float*)d_out;

  geoflow_kernel<<<dim3(BATCH), dim3(256), 0, stream>>>(
      points, Wm1, bm1, Wm2, bm2, Wc1, bc1, Wc2, bc2,
      Wr1, br1, Wr2, br2, Wf1, bf1, Wf2, bf2, outp);
}